// IndexedConv_13219909337240
// MI455X (gfx1250) — compile-verified
//
#include <hip/hip_runtime.h>

typedef float v2f __attribute__((ext_vector_type(2)));
typedef float v8f __attribute__((ext_vector_type(8)));

#define N_BATCH 16
#define C_IN    32
#define C_OUT   128
#define KSZ     7
#define LEN     40000
#define RED     (C_IN * KSZ)   // 224
#define NSTEP   (RED / 4)      // 56 k-steps of 4
#define MT      (C_OUT / 16)   // 8 output row tiles
#define WAVES   4              // waves (L tiles) per block
#define NGRP    4              // batch images per block
#define SENT    0xFFFFFFFFu

// ---------------------------------------------------------------------------
// Prep: rearrange weight (C_OUT, C_IN*K) row-major into per-lane WMMA A-
// fragment order in d_ws.  Fragment (mt, s): lane l holds
//   a[j] = w[o = mt*16 + (l%16)][r = 4*s + 2*(l/16) + j],  j = 0,1
// stored as v2f at wsA[(mt*NSTEP + s)*32 + l].
// ---------------------------------------------------------------------------
__global__ void prep_weights(const float* __restrict__ w, v2f* __restrict__ wsA) {
    int t = blockIdx.x * blockDim.x + threadIdx.x;
    if (t >= MT * NSTEP * 32) return;
    int lane = t & 31;
    int ms   = t >> 5;
    int s    = ms % NSTEP;
    int mt   = ms / NSTEP;
    int o    = mt * 16 + (lane & 15);
    int r0   = 4 * s + 2 * (lane >> 4);
    v2f a;
    a[0] = w[o * RED + r0];
    a[1] = w[o * RED + r0 + 1];
    wsA[t] = a;
}

// ---------------------------------------------------------------------------
// Main: each wave computes a 128(C_out) x 16(L) tile for NGRP batch images.
// x (82 MB) is L2-resident and re-gathered ~7x; the 328 MB output stream is
// written with non-temporal hints so it does not evict x from the 192 MB L2.
// ---------------------------------------------------------------------------
__global__ void __launch_bounds__(WAVES * 32) idxconv_main(
    const float* __restrict__ x,     // (N, C_IN, LEN)
    const v2f*   __restrict__ wsA,   // rearranged weights
    const float* __restrict__ bias,  // (C_OUT)
    const int*   __restrict__ idx,   // (KSZ, LEN)
    float*       __restrict__ out)   // (N, C_OUT, LEN)
{
    // Per-wave resolved gather offsets, stored as (even r, odd r) uint pairs:
    // soff[wave][ (p*16 + col)*2 + j ] = element offset into x[n] plane,
    // p = r>>1, j = r&1; SENT marks masked (idx == -1) entries.
    __shared__ unsigned int soff[WAVES][NSTEP * 2 * 16];   // 112*16 uints/wave
    __shared__ __align__(16) float sbias[C_OUT];

    const int tid  = threadIdx.x;
    const int wave = tid >> 5;
    const int lane = tid & 31;
    const int col  = lane & 15;   // L column within tile / WMMA N index
    const int h    = lane >> 4;   // lane half

    if (tid < C_OUT) sbias[tid] = bias[tid];

    const int l0 = (blockIdx.x * WAVES + wave) * 16;   // 2500 tiles exactly

    // --- resolve gather offsets for this L tile (shared across NGRP batches)
    unsigned int* so = &soff[wave][0];
    for (int e = lane; e < (RED / 2) * 16; e += 32) {
        int p  = e >> 4;          // r pair index
        int cc = e & 15;          // column
        int r0 = 2 * p;
        int r1 = r0 + 1;
        int i0 = __builtin_nontemporal_load(&idx[(r0 % KSZ) * LEN + l0 + cc]);
        int i1 = __builtin_nontemporal_load(&idx[(r1 % KSZ) * LEN + l0 + cc]);
        so[2 * e + 0] = (i0 < 0) ? SENT : (unsigned int)((r0 / KSZ) * LEN + i0);
        so[2 * e + 1] = (i1 < 0) ? SENT : (unsigned int)((r1 / KSZ) * LEN + i1);
    }
    __syncthreads();

    for (int nn = 0; nn < NGRP; ++nn) {
        const int n = blockIdx.y * NGRP + nn;
        const float* __restrict__ xn = x + (size_t)n * C_IN * LEN;

        v8f acc[MT] = {};

        for (int s = 0; s < NSTEP; ++s) {
            // B fragment: lane l -> B[K = 2h + j][Ncol = col], r = 4s + 2h + j
            const unsigned int* op = &so[((2 * s + h) * 16 + col) * 2];
            unsigned int o0 = op[0];
            unsigned int o1 = op[1];
            bool m0 = (o0 != SENT);
            bool m1 = (o1 != SENT);
            float b0 = xn[m0 ? o0 : 0u];
            float b1 = xn[m1 ? o1 : 0u];
            v2f b;
            b[0] = m0 ? b0 : 0.0f;
            b[1] = m1 ? b1 : 0.0f;

            #pragma unroll
            for (int m = 0; m < MT; ++m) {
                v2f a = wsA[(m * NSTEP + s) * 32 + lane];
                acc[m] = __builtin_amdgcn_wmma_f32_16x16x4_f32(
                    /*neg_a=*/false, a, /*neg_b=*/false, b,
                    /*c_mod=*/(short)0, acc[m],
                    /*reuse_a=*/false, /*reuse_b=*/false);
            }
        }

        // --- bias + store.  D layout: VGPR v, lane l -> M = v + 8h, N = col.
        // Non-temporal stores: keep the 328 MB output stream out of L2 so the
        // gather working set (x) stays resident.
        #pragma unroll
        for (int m = 0; m < MT; ++m) {
            const int ob = m * 16 + 8 * h;
            const float4* bq = (const float4*)&sbias[ob];
            float4 bb0 = bq[0];
            float4 bb1 = bq[1];
            float bv[8] = {bb0.x, bb0.y, bb0.z, bb0.w, bb1.x, bb1.y, bb1.z, bb1.w};
            float* __restrict__ op = out + ((size_t)n * C_OUT + ob) * LEN + l0 + col;
            #pragma unroll
            for (int v = 0; v < 8; ++v) {
                __builtin_nontemporal_store(acc[m][v] + bv[v], &op[(size_t)v * LEN]);
            }
        }
    }
}

extern "C" void kernel_launch(void* const* d_in, const int* in_sizes, int n_in,
                              void* d_out, int out_size, void* d_ws, size_t ws_size,
                              hipStream_t stream) {
    const float* x    = (const float*)d_in[0];
    const float* w    = (const float*)d_in[1];
    const float* bias = (const float*)d_in[2];
    const int*   idx  = (const int*)d_in[3];
    float*       out  = (float*)d_out;
    v2f*         wsA  = (v2f*)d_ws;   // 14336 * 8B = 115 KB of scratch

    prep_weights<<<(MT * NSTEP * 32 + 255) / 256, 256, 0, stream>>>(w, wsA);

    dim3 grid((LEN / 16) / WAVES, N_BATCH / NGRP);   // (625, 4)
    idxconv_main<<<grid, WAVES * 32, 0, stream>>>(x, wsA, bias, idx, out);
}